// QuantumNATExtended_65481071409906
// MI455X (gfx1250) — compile-verified
//
#include <hip/hip_runtime.h>
#include <math.h>

// ---------------------------------------------------------------------------
// QuantumNAT fused pipeline for gfx1250 (MI455X).
// Kernel 1: stream x (302 MB, the roofline term) -> pool -> encode -> 4-qubit
//           statevector sim in VGPRs -> WMMA f32 16x16x4 MLP -> pre-BN out +
//           deterministic per-block channel sums.
// Kernel 2: deterministic reduce of per-block partials -> BN scale/bias.
// Kernel 3: in-place affine on d_out.
// ---------------------------------------------------------------------------

typedef __attribute__((ext_vector_type(2))) float v2f;
typedef __attribute__((ext_vector_type(8))) float v8f;

__device__ __forceinline__ v8f wmma_16x16x4_f32(v2f a, v2f b, v8f c) {
  // 8-arg form: (neg_a, A, neg_b, B, c_mod, C, reuse_a, reuse_b)
  return __builtin_amdgcn_wmma_f32_16x16x4_f32(
      false, a, false, b, (short)0, c, false, false);
}

struct cplx { float x, y; };
__device__ __forceinline__ cplx cmul(cplx a, cplx b) {
  cplx r; r.x = a.x * b.x - a.y * b.y; r.y = a.x * b.y + a.y * b.x; return r;
}

__global__ __launch_bounds__(256) void qnat_main_kernel(
    const float* __restrict__ x, const float* __restrict__ W_enc,
    const float* __restrict__ b_enc, const float* __restrict__ W1,
    const float* __restrict__ b1, const float* __restrict__ W2,
    const float* __restrict__ b2, float* __restrict__ out,
    float* __restrict__ partials, int Btot) {
  __shared__ float4 qbuf[256];       // out_q per sample
  __shared__ float  hbuf[8][16][33]; // per-wave 16x32 hidden tile (+pad)
  __shared__ float  redS[4][32];     // per-channel partial sums
  __shared__ float  redQ[4][32];     // per-channel partial sumsq

  const int tid = threadIdx.x;
  const int s   = blockIdx.x * 256 + tid;

  // ---------------- Stage 1: per-lane sample pipeline --------------------
  float ez0 = 0.f, ez1 = 0.f, ez2 = 0.f, ez3 = 0.f;
  if (s < Btot) {
    // 6x6 average pool of 24x24 -> pooled[16], fully unrolled (regs only)
    float pooled[16];
#pragma unroll
    for (int i = 0; i < 16; ++i) pooled[i] = 0.f;
    const float* xp = x + (size_t)s * 576;
#pragma unroll
    for (int r = 0; r < 24; ++r) {
      const float4* rp = (const float4*)(xp + r * 24);
      float4 v0 = rp[0], v1 = rp[1], v2 = rp[2];
      float4 v3 = rp[3], v4 = rp[4], v5 = rp[5];
      const int pr = (r / 6) * 4;
      pooled[pr + 0] += v0.x + v0.y + v0.z + v0.w + v1.x + v1.y;
      pooled[pr + 1] += v1.z + v1.w + v2.x + v2.y + v2.z + v2.w;
      pooled[pr + 2] += v3.x + v3.y + v3.z + v3.w + v4.x + v4.y;
      pooled[pr + 3] += v4.z + v4.w + v5.x + v5.y + v5.z + v5.w;
    }
#pragma unroll
    for (int i = 0; i < 16; ++i) pooled[i] *= (1.f / 36.f);

    // encode: enc[o] = b_enc[o] + sum_k pooled[k] * W_enc[o,k]
    float enc[4];
#pragma unroll
    for (int o = 0; o < 4; ++o) {
      float acc = b_enc[o];
#pragma unroll
      for (int k = 0; k < 16; ++k) acc += pooled[k] * W_enc[o * 16 + k];
      enc[o] = acc;
    }

    // per-qubit gate scalars
    float ch[4], sh[4];
#pragma unroll
    for (int w = 0; w < 4; ++w) {
      float h = 0.5f * enc[w];
      ch[w] = __cosf(h);
      sh[w] = __sinf(h);
    }
    // after RY then RZ on |0>:  alpha=(c^2,-cs), beta=(cs,s^2)
    cplx g0[4], g1[4];
#pragma unroll
    for (int w = 0; w < 4; ++w) {
      g0[w].x = ch[w] * ch[w]; g0[w].y = -ch[w] * sh[w];
      g1[w].x = ch[w] * sh[w]; g1[w].y =  sh[w] * sh[w];
    }
    // product state -> 16 amplitudes (n = q0*8 + q1*4 + q2*2 + q3)
    cplx t01[4], t23[4], psi[16];
#pragma unroll
    for (int a = 0; a < 2; ++a)
#pragma unroll
      for (int b = 0; b < 2; ++b) {
        t01[a * 2 + b] = cmul(a ? g1[0] : g0[0], b ? g1[1] : g0[1]);
        t23[a * 2 + b] = cmul(a ? g1[2] : g0[2], b ? g1[3] : g0[3]);
      }
#pragma unroll
    for (int i = 0; i < 4; ++i)
#pragma unroll
      for (int j = 0; j < 4; ++j) psi[i * 4 + j] = cmul(t01[i], t23[j]);

    // CNOT chain (pure index permutation -> register swaps)
#define QSW(a, b) { cplx t = psi[a]; psi[a] = psi[b]; psi[b] = t; }
    QSW(8, 12)  QSW(9, 13)  QSW(10, 14) QSW(11, 15)   // CNOT(0,1)
    QSW(4, 6)   QSW(5, 7)   QSW(12, 14) QSW(13, 15)   // CNOT(1,2)
    QSW(2, 3)   QSW(6, 7)   QSW(10, 11) QSW(14, 15)   // CNOT(2,3)
#undef QSW

    // RX on each wire
#pragma unroll
    for (int w = 0; w < 4; ++w) {
      const int bv = 8 >> w;
      const float c = ch[w], sn = sh[w];
#pragma unroll
      for (int n = 0; n < 16; ++n) {
        if ((n & bv) == 0) {
          cplx p0 = psi[n], p1 = psi[n | bv];
          psi[n].x      = c * p0.x + sn * p1.y;
          psi[n].y      = c * p0.y - sn * p1.x;
          psi[n | bv].x = sn * p0.y + c * p1.x;
          psi[n | bv].y = c * p1.y - sn * p0.x;
        }
      }
    }

    // PauliZ expectations
    float p[16];
#pragma unroll
    for (int n = 0; n < 16; ++n) p[n] = psi[n].x * psi[n].x + psi[n].y * psi[n].y;
#pragma unroll
    for (int n = 0; n < 16; ++n) {
      ez0 += (n & 8) ? -p[n] : p[n];
      ez1 += (n & 4) ? -p[n] : p[n];
      ez2 += (n & 2) ? -p[n] : p[n];
      ez3 += (n & 1) ? -p[n] : p[n];
    }
  }
  qbuf[tid] = make_float4(ez0, ez1, ez2, ez3);
  __syncthreads();

  // ---------------- Stage 2: WMMA MLP over 16-sample tiles ---------------
  const int lane = tid & 31;
  const int wv   = tid >> 5;   // wave id (8 waves)
  const int mrow = lane & 15;  // M row / N col index
  const int hi   = lane >> 4;  // A-matrix K half: 0 -> K{0,1}, 1 -> K{2,3}
  const int jc   = mrow & 3;   // clamped class index (cols 4..15 of D unread)

#pragma unroll
  for (int t2 = 0; t2 < 2; ++t2) {
    const int tile = wv * 2 + t2;

    // A1: out_q tile, A-layout for 16x16x4
    float4 q = qbuf[tile * 16 + mrow];
    v2f a1;
    if (hi) { a1.x = q.z; a1.y = q.w; } else { a1.x = q.x; a1.y = q.y; }

    // Layer 1: h[16,32] = relu(A1 @ W1^T + b1), two 16-wide N halves
#pragma unroll
    for (int n = 0; n < 2; ++n) {
      const int j = n * 16 + mrow;           // hidden unit (N)
      v2f b;                                  // B[K][N] = W1[j][K]
      b.x = W1[j * 4 + 2 * hi];
      b.y = W1[j * 4 + 2 * hi + 1];
      const float bb = b1[j];                 // bias broadcast along M
      v8f cc = {bb, bb, bb, bb, bb, bb, bb, bb};
      v8f hh = wmma_16x16x4_f32(a1, b, cc);
#pragma unroll
      for (int i = 0; i < 8; ++i) {           // D: VGPR i -> rows i, i+8
        const float hv = hh[i];
        hbuf[wv][i + 8 * hi][j] = (hv > 0.f) ? hv : 0.f;
      }
    }
    // wave-private LDS region: same-wave DS ordering + compiler dscnt waits

    // Layer 2: out[16,4] = h @ W2^T + b2, 8 accumulating K=4 chunks.
    // B columns N=4..15 carry clamped-index garbage; D columns 4..15 are
    // never read, and WMMA dot products do not mix N columns, so this is
    // safe and avoids divergent per-element loads.
    const float ci = b2[jc];
    v8f acc = {ci, ci, ci, ci, ci, ci, ci, ci};
#pragma unroll
    for (int kc = 0; kc < 8; ++kc) {
      v2f a2;                                 // A: h[sample=mrow][k-chunk]
      a2.x = hbuf[wv][mrow][4 * kc + 2 * hi];
      a2.y = hbuf[wv][mrow][4 * kc + 2 * hi + 1];
      v2f bv;                                 // B[K][N] = W2[class][K]
      bv.x = W2[jc * 32 + 4 * kc + 2 * hi];
      bv.y = W2[jc * 32 + 4 * kc + 2 * hi + 1];
      acc = wmma_16x16x4_f32(a2, bv, acc);
    }

    // scatter pre-BN results + per-channel partial sums (valid cols only)
    if (mrow < 4) {
      const int sBase = blockIdx.x * 256 + tile * 16 + 8 * hi;
      float s1 = 0.f, s2 = 0.f;
#pragma unroll
      for (int i = 0; i < 8; ++i) {
        const float v = acc[i];
        const int srow = sBase + i;
        if (srow < Btot) out[(size_t)srow * 4 + mrow] = v;
        s1 += v; s2 += v * v;
      }
      redS[mrow][tile * 2 + hi] = s1;
      redQ[mrow][tile * 2 + hi] = s2;
    }
  }
  __syncthreads();

  // deterministic fixed-order block reduction of the 32 partials per channel
  if (tid < 8) {
    const int j = tid & 3;
    const bool sq = tid >= 4;
    float tot = 0.f;
#pragma unroll
    for (int pp = 0; pp < 32; ++pp) tot += sq ? redQ[j][pp] : redS[j][pp];
    partials[(size_t)blockIdx.x * 8 + (sq ? 4 : 0) + j] = tot;
  }
}

__global__ void qnat_bnstats_kernel(const float* __restrict__ partials,
                                    int nblocks, int Btot,
                                    const float* __restrict__ gamma,
                                    const float* __restrict__ beta,
                                    float* __restrict__ sb) {
  __shared__ float tots[8];
  const int t = threadIdx.x;
  if (t < 8) {
    const int j = t & 3;
    const int off = (t >= 4) ? 4 : 0;
    float tot = 0.f;
    for (int b = 0; b < nblocks; ++b) tot += partials[(size_t)b * 8 + off + j];
    tots[t] = tot;
  }
  __syncthreads();
  if (t < 4) {
    const float invB = 1.f / (float)Btot;
    const float mean = tots[t] * invB;
    float var = tots[4 + t] * invB - mean * mean;
    var = fmaxf(var, 0.f);
    const float sc = gamma[t] / sqrtf(var + 1e-5f);
    sb[t] = sc;
    sb[4 + t] = beta[t] - mean * sc;
  }
}

__global__ __launch_bounds__(256) void qnat_bnapply_kernel(
    float* __restrict__ out, const float* __restrict__ sb, int Btot) {
  const int s = blockIdx.x * blockDim.x + threadIdx.x;
  if (s >= Btot) return;
  const float4 sc = ((const float4*)sb)[0];
  const float4 bi = ((const float4*)sb)[1];
  float4* o4 = (float4*)out;
  float4 v = o4[s];
  v.x = v.x * sc.x + bi.x;
  v.y = v.y * sc.y + bi.y;
  v.z = v.z * sc.z + bi.z;
  v.w = v.w * sc.w + bi.w;
  o4[s] = v;
}

extern "C" void kernel_launch(void* const* d_in, const int* in_sizes, int n_in,
                              void* d_out, int out_size, void* d_ws,
                              size_t ws_size, hipStream_t stream) {
  const float* x     = (const float*)d_in[0];
  const float* W_enc = (const float*)d_in[1];
  const float* b_enc = (const float*)d_in[2];
  const float* W1    = (const float*)d_in[3];
  const float* b1    = (const float*)d_in[4];
  const float* W2    = (const float*)d_in[5];
  const float* b2    = (const float*)d_in[6];
  const float* gamma = (const float*)d_in[7];
  const float* beta  = (const float*)d_in[8];

  const int Btot    = in_sizes[0] / 576;       // [B,1,24,24]
  const int nblocks = (Btot + 255) / 256;

  float* out      = (float*)d_out;
  float* partials = (float*)d_ws;              // nblocks*8 floats
  float* sb       = partials + (size_t)nblocks * 8;  // 8 floats (16B aligned)

  qnat_main_kernel<<<nblocks, 256, 0, stream>>>(x, W_enc, b_enc, W1, b1, W2,
                                                b2, out, partials, Btot);
  qnat_bnstats_kernel<<<1, 64, 0, stream>>>(partials, nblocks, Btot, gamma,
                                            beta, sb);
  qnat_bnapply_kernel<<<(Btot + 255) / 256, 256, 0, stream>>>(out, sb, Btot);
}